// BezierAlign_34402688040966
// MI455X (gfx1250) — compile-verified
//
#include <hip/hip_runtime.h>

typedef float v2f __attribute__((ext_vector_type(2)));
typedef float v8f __attribute__((ext_vector_type(8)));

static constexpr int BATCH = 8, CH = 256, H = 128, W = 128;
static constexpr int NROI = 256, PH = 16, PW = 64;
static constexpr float SCALE = 0.25f;

// ---------------------------------------------------------------------------
// Kernel 1: Bezier endpoint evaluation as a batched (16x4)x(4x16) matmul on
// the tensor path.  x_top/y_top/x_bot/y_bot for 256 curves x 64 u-samples.
//   A (16x4, MxK) f32: lanes 0-15 = curve rows, VGPR0/1 = K{0,1}; lanes 16-31
//                      hold K{2,3}.
//   B (4x16, KxN) f32: lane = u column, VGPR0/1 = K{0,1}; lanes 16-31 = K{2,3}.
//   D (16x16) f32: 8 VGPRs, lane<16 -> M=r, lane>=16 -> M=r+8.
// One wave per 16-curve tile; EXEC all-ones (32 threads/block).
// ---------------------------------------------------------------------------
__global__ void __launch_bounds__(32)
bezier_coords_wmma(const float* __restrict__ bez, float* __restrict__ coords)
{
    const int lane = threadIdx.x;      // 0..31
    const int half = lane >> 4;        // K group: 0 -> K{0,1}, 1 -> K{2,3}
    const int lm   = lane & 15;
    const int ntile = blockIdx.x;      // 16 tiles of 16 curves
    const int n_a   = ntile * 16 + lm; // curve index feeding the A rows

    const int k0 = half * 2;
    const float* arow = bez + n_a * 17;

    #pragma unroll
    for (int arr = 0; arr < 4; ++arr) {
        // bezier row = [batch, c0x,c0y, c1x,c1y, ...]:
        // x_top cols {1,3,5,7}, y_top {2,4,6,8}, x_bot {9,11,13,15}, y_bot {10,12,14,16}
        const int colbase = (arr < 2 ? 1 : 9) + (arr & 1);
        v2f a;
        a.x = arow[colbase + 2 * (k0 + 0)] * SCALE;
        a.y = arow[colbase + 2 * (k0 + 1)] * SCALE;

        #pragma unroll
        for (int ut = 0; ut < 4; ++ut) {
            const float t  = (float)(ut * 16 + lm) * (1.0f / PW);
            const float mt = 1.0f - t;
            const float bern0 = mt * mt * mt;
            const float bern1 = 3.0f * t * mt * mt;
            const float bern2 = 3.0f * t * t * mt;
            const float bern3 = t * t * t;
            v2f bv;
            bv.x = (k0 == 0) ? bern0 : bern2;
            bv.y = (k0 == 0) ? bern1 : bern3;

            v8f c = {};
            v8f d = __builtin_amdgcn_wmma_f32_16x16x4_f32(
                        /*neg_a=*/false, a, /*neg_b=*/false, bv,
                        /*c_mod=*/(short)0, c, /*reuse_a=*/false, /*reuse_b=*/false);

            const int u_out = ut * 16 + lm;
            const int nbase = ntile * 16 + half * 8;
            #pragma unroll
            for (int r = 0; r < 8; ++r)
                coords[arr * (NROI * PW) + (nbase + r) * PW + u_out] = d[r];
        }
    }
}

// ---------------------------------------------------------------------------
// Kernel 2: the gather-bound BezierAlign proper.
// Block = 256 threads = 64 px-lanes x 4 py rows; grid = 256 curves x 4 y-tiles.
// All address/weight math hoisted out of the 256-iteration channel loop.
// Output streamed with non-temporal stores (write-once 256 MB) so the 128 MB
// input stays resident in the 192 MB L2.
// ---------------------------------------------------------------------------
__global__ void __launch_bounds__(256)
bezier_align_kernel(const float* __restrict__ input, const float* __restrict__ bez,
                    const float* __restrict__ coords, float* __restrict__ out)
{
    const int tid = threadIdx.x;
    const int px  = tid & (PW - 1);
    const int pyl = tid >> 6;                 // 0..3
    const int n   = blockIdx.x >> 2;
    const int py  = (blockIdx.x & 3) * 4 + pyl;

    const float* brow = bez + n * 17;
    const int b = (int)brow[0];

    float xt, yt, xb, yb;
    if (coords) {                              // WMMA-precomputed endpoints
        const int o = n * PW + px;
        xt = coords[0 * (NROI * PW) + o];
        yt = coords[1 * (NROI * PW) + o];
        xb = coords[2 * (NROI * PW) + o];
        yb = coords[3 * (NROI * PW) + o];
    } else {                                   // fallback: inline VALU eval
        const float t = (float)px * (1.0f / PW), mt = 1.0f - t;
        const float b0 = mt * mt * mt, b1 = 3.f * t * mt * mt,
                    b2 = 3.f * t * t * mt, b3 = t * t * t;
        xt = (brow[1] * b0 + brow[3]  * b1 + brow[5]  * b2 + brow[7]  * b3) * SCALE;
        yt = (brow[2] * b0 + brow[4]  * b1 + brow[6]  * b2 + brow[8]  * b3) * SCALE;
        xb = (brow[9] * b0 + brow[11] * b1 + brow[13] * b2 + brow[15] * b3) * SCALE;
        yb = (brow[10]* b0 + brow[12] * b1 + brow[14] * b2 + brow[16] * b3) * SCALE;
    }

    const float v = (float)py * (1.0f / PH);
    const float x = xb * v + xt * (1.0f - v);
    const float y = yb * v + yt * (1.0f - v);

    const bool valid = (y > -1.0f) && (y < (float)H) && (x > -1.0f) && (x < (float)W);
    const float xs = fmaxf(x, 0.0f), ys = fmaxf(y, 0.0f);
    const int xl = min((int)floorf(xs), W - 1);
    const int yl = min((int)floorf(ys), H - 1);
    const int xh = min(xl + 1, W - 1);
    const int yh = min(yl + 1, H - 1);
    const float lx = (xl >= W - 1) ? 0.0f : (xs - (float)xl);
    const float ly = (yl >= H - 1) ? 0.0f : (ys - (float)yl);
    const float hx = 1.0f - lx, hy = 1.0f - ly;
    float w00 = hy * hx, w01 = hy * lx, w10 = ly * hx, w11 = ly * lx;
    if (!valid) { w00 = w01 = w10 = w11 = 0.0f; }

    const int o00 = yl * W + xl, o01 = yl * W + xh;
    const int o10 = yh * W + xl, o11 = yh * W + xh;
    const float* p = input + b * (CH * H * W);
    float* optr = out + n * (CH * PH * PW) + py * PW + px;

    #pragma unroll 4
    for (int c = 0; c < CH; ++c) {
        const int cb = c * (H * W);
        const float v00 = p[cb + o00];
        const float v01 = p[cb + o01];
        const float v10 = p[cb + o10];
        const float v11 = p[cb + o11];
        const float r = w00 * v00 + w01 * v01 + w10 * v10 + w11 * v11;
        __builtin_nontemporal_store(r, optr + c * (PH * PW));
    }
}

extern "C" void kernel_launch(void* const* d_in, const int* in_sizes, int n_in,
                              void* d_out, int out_size, void* d_ws, size_t ws_size,
                              hipStream_t stream)
{
    const float* input = (const float*)d_in[0];
    const float* bez   = (const float*)d_in[1];
    float* out    = (float*)d_out;
    float* coords = (float*)d_ws;

    const size_t need = (size_t)4 * NROI * PW * sizeof(float);   // 256 KB
    const bool use_wmma = (d_ws != nullptr) && (ws_size >= need);

    if (use_wmma)
        bezier_coords_wmma<<<NROI / 16, 32, 0, stream>>>(bez, coords);

    bezier_align_kernel<<<NROI * (PH / 4), 256, 0, stream>>>(
        input, bez, use_wmma ? coords : nullptr, out);
}